// ToHeteroLinear_19421842113014
// MI455X (gfx1250) — compile-verified
//
#include <hip/hip_runtime.h>

typedef __attribute__((ext_vector_type(2))) float v2f;
typedef __attribute__((ext_vector_type(4))) float v4f;
typedef __attribute__((ext_vector_type(8))) float v8f;

static constexpr int D = 128;            // D_IN == D_OUT == 128
static constexpr int ROWS_PER_BLOCK = 256;
static constexpr int ROWS_PER_WAVE = 32; // two 16-row WMMA tiles per wave

#define WMMA_F32(A, B, C) \
  __builtin_amdgcn_wmma_f32_16x16x4_f32(false, (A), false, (B), (short)0, (C), false, false)

__global__ __launch_bounds__(256) void hetero_linear_wmma(
    const float* __restrict__ x,
    const int*   __restrict__ tvec,
    const float* __restrict__ W,
    const float* __restrict__ bias,
    float*       __restrict__ out,
    int nrows)
{
  // W[t] staged pair-row interleaved: element (k,c) -> lds[(k>>1)*256 + (c ^ (((k>>1)&1)<<4))*2 + (k&1)]
  // => a 16x16x4 B fragment (rows k,k+1, one column) is one contiguous 8B ds_load_b64,
  //    and the XOR-16 column swizzle makes the two half-waves hit disjoint bank halves.
  __shared__ float ldsW[64 * 256];   // 64 KB

  const int tid       = threadIdx.x;
  const int block_row = blockIdx.x * ROWS_PER_BLOCK;

  // Rows sorted by type: equal endpoints => uniform type across the block.
  const int t_first = tvec[block_row];
  const int t_last  = tvec[block_row + ROWS_PER_BLOCK - 1];

  if (t_first == t_last) {
    // ---------------- fast path: single type, WMMA f32 ----------------
    const float* Wt = W + (size_t)t_first * D * D;

    // Stage W[t]: 64 pair-rows x 32 col-blocks (4 cols) = 2048 tasks / 256 threads.
    for (int task = tid; task < 64 * 32; task += 256) {
      const int pr = task >> 5;              // pair-row (K rows 2pr, 2pr+1)
      const int cb = (task & 31) << 2;       // column base
      v4f g0 = *(const v4f*)(Wt + (size_t)(2 * pr)     * D + cb);
      v4f g1 = *(const v4f*)(Wt + (size_t)(2 * pr + 1) * D + cb);
      const int cs = cb ^ ((pr & 1) << 4);   // XOR-16 column swizzle
      float* dst = &ldsW[pr * 256 + cs * 2];
      v4f lo = { g0[0], g1[0], g0[1], g1[1] };
      v4f hi = { g0[2], g1[2], g0[3], g1[3] };
      *(v4f*)(dst)     = lo;
      *(v4f*)(dst + 4) = hi;
    }
    __syncthreads();

    const int lane = tid & 31;
    const int wave = tid >> 5;
    const int h    = lane >> 4;              // half-wave select
    const int l    = lane & 15;

    const int r0 = block_row + wave * ROWS_PER_WAVE;   // row tiles at r0, r0+16

    v8f acc[2][8];
    #pragma unroll
    for (int rt = 0; rt < 2; ++rt)
      #pragma unroll
      for (int j = 0; j < 8; ++j)
        acc[rt][j] = v8f{0.f, 0.f, 0.f, 0.f, 0.f, 0.f, 0.f, 0.f};

    const float* xr0 = x + (size_t)(r0 + l)      * D;
    const float* xr1 = x + (size_t)(r0 + 16 + l) * D;

    // Per-lane B base (pair-row h, lane column l); swizzle folds to h<<4.
    // Column tile j lives at (j>>1)*64 + ((j&1)^h)*32 floats from here.
    const float* bLane = ldsW + h * 256 + (l << 1);
    const float* bE;   // even j
    const float* bO;   // odd j
    {
      const int offE = h * 32;
      bE = bLane + offE;
      bO = bLane + (32 - offE);
    }

    for (int k0 = 0; k0 < D; k0 += 4) {
      // A fragments (16x4 f32): lane row = l, K = k0 + 2h + {0,1}; 8B global load.
      v2f a0 = *(const v2f*)(xr0 + k0 + 2 * h);
      v2f a1 = *(const v2f*)(xr1 + k0 + 2 * h);
      const int kOff = (k0 >> 1) * 256;      // pair-row advance

      // Hoist all 8 B fragments first so the 4 ds_load_2addr_b64 issue
      // back-to-back and the dscnt waits amortize across the WMMA burst.
      v2f bfrag[8];
      #pragma unroll
      for (int j = 0; j < 8; ++j) {
        const float* bp = ((j & 1) ? bO : bE) + kOff + (j >> 1) * 64;
        bfrag[j] = *(const v2f*)bp;          // one ds_load_b64 = full B fragment
      }
      #pragma unroll
      for (int j = 0; j < 8; ++j) {
        acc[0][j] = WMMA_F32(a0, bfrag[j], acc[0][j]);
        acc[1][j] = WMMA_F32(a1, bfrag[j], acc[1][j]);
      }
    }

    // Store: acc[rt][j][i] -> row r0 + rt*16 + 8h + i, col j*16 + l ; add bias.
    #pragma unroll
    for (int j = 0; j < 8; ++j) {
      const int   col = j * 16 + l;
      const float bv  = bias[t_first * D + col];
      #pragma unroll
      for (int rt = 0; rt < 2; ++rt) {
        float* orow = out + (size_t)(r0 + rt * 16 + 8 * h) * D + col;
        #pragma unroll
        for (int i = 0; i < 8; ++i)
          orow[(size_t)i * D] = acc[rt][j][i] + bv;
      }
    }
  } else {
    // ------------- boundary blocks (<=15 of 512): per-row VALU fallback -------------
    for (int idx = tid; idx < ROWS_PER_BLOCK * D; idx += 256) {
      const int r = block_row + (idx >> 7);
      const int c = idx & (D - 1);
      if (r >= nrows) break;
      const int t = tvec[r];
      const float* Wt = W + (size_t)t * D * D + c;
      const float* xr = x + (size_t)r * D;
      float a = bias[t * D + c];
      #pragma unroll 8
      for (int k = 0; k < D; ++k)
        a = fmaf(xr[k], Wt[(size_t)k * D], a);
      out[(size_t)r * D + c] = a;
    }
  }
}

extern "C" void kernel_launch(void* const* d_in, const int* in_sizes, int n_in,
                              void* d_out, int out_size, void* d_ws, size_t ws_size,
                              hipStream_t stream) {
  const float* x    = (const float*)d_in[0];
  const int*   tvec = (const int*)  d_in[1];
  const float* W    = (const float*)d_in[2];
  const float* bias = (const float*)d_in[3];
  float* out = (float*)d_out;

  const int nrows = in_sizes[0] / D;           // 131072
  const int grid  = nrows / ROWS_PER_BLOCK;    // 512 blocks of 256 threads (8 waves)

  hipLaunchKernelGGL(hetero_linear_wmma, dim3(grid), dim3(256), 0, stream,
                     x, tvec, W, bias, out, nrows);
}